// SimpleGAT_89266600280694
// MI455X (gfx1250) — compile-verified
//
#include <hip/hip_runtime.h>
#include <math.h>

#define N_NODES  50000
#define E_ORIG   800000
#define E_TOT    850000
#define HEADS    4
#define N_GRAPHS 64
#define NEG_SLOPE 0.2f
#define GN_EPS    1e-5f

typedef float v2f __attribute__((ext_vector_type(2)));
typedef float v8f __attribute__((ext_vector_type(8)));

// ------------------------------------------------------------------
// utility: zero a buffer (works for f32 zeros and u32 sentinels)
// ------------------------------------------------------------------
__global__ void zero_u32_kernel(unsigned* __restrict__ p, long n) {
  long i = (long)blockIdx.x * blockDim.x + threadIdx.x;
  if (i < n) p[i] = 0u;
}

static inline unsigned gdim(long n, int b) { return (unsigned)((n + b - 1) / b); }

static inline void zero_buf(void* p, long n, hipStream_t s) {
  zero_u32_kernel<<<gdim(n, 256), 256, 0, s>>>((unsigned*)p, n);
}

// ------------------------------------------------------------------
// fp32 WMMA GEMM: C[M,N] = A[M,K] @ B[K,N], all row-major.
// M % 16 == 0, N % 16 == 0, K % 4 == 0. One wave per 16x16 C tile,
// V_WMMA_F32_16X16X4_F32 stepping K by 4.
// A frag (16x4): lane L -> row L%16; VGPR j -> K = k0 + 2*(L/16) + j
// B frag (4x16): lane L -> col L%16; VGPR j -> K = k0 + 2*(L/16) + j
// C/D (16x16):   lane L -> col L%16; VGPR r -> row r + 8*(L/16)
// ------------------------------------------------------------------
__global__ void gemm_wmma_f32(const float* __restrict__ A, const float* __restrict__ B,
                              float* __restrict__ C, int M, int N, int K) {
  int wv   = (int)((blockIdx.x * blockDim.x + threadIdx.x) >> 5);
  int lane = threadIdx.x & 31;
  int mtiles = M >> 4, ntiles = N >> 4;
  if (wv >= mtiles * ntiles) return;   // uniform per wave -> EXEC all-1s for WMMA
  int mt = wv / ntiles, nt = wv % ntiles;
  int r    = lane & 15;   // row-in-tile for A, col-in-tile for B/C
  int half = lane >> 4;   // 0 or 1

  const float* arow = A + (long)(mt * 16 + r) * K;
  const float* bcol = B + (nt * 16 + r);
  v8f acc = {};
  for (int k0 = 0; k0 < K; k0 += 4) {
    int ka = k0 + 2 * half;
    v2f a, b;
    a[0] = arow[ka];
    a[1] = arow[ka + 1];
    b[0] = bcol[(long)ka * N];
    b[1] = bcol[(long)(ka + 1) * N];
    acc = __builtin_amdgcn_wmma_f32_16x16x4_f32(false, a, false, b,
                                                (short)0, acc, false, false);
  }
  float* crow = C + (long)(mt * 16 + half * 8) * N + nt * 16 + r;
  for (int i = 0; i < 8; ++i) crow[(long)i * N] = acc[i];
}

// ------------------------------------------------------------------
// edge helpers
// ------------------------------------------------------------------
__device__ __forceinline__ void edge_endpoints(const int* __restrict__ eidx,
                                               int e, int& s, int& d) {
  if (e < E_ORIG) { s = eidx[e]; d = eidx[E_ORIG + e]; }
  else            { s = d = e - E_ORIG; }      // self loops appended
}

// per (edge, head): e = att[h] . leaky_relu(xl[src] + xr[dst]); seg-max into emax
__global__ void edge_scores_kernel(const float* __restrict__ xl, const float* __restrict__ xr,
                                   const float* __restrict__ att, const int* __restrict__ eidx,
                                   float* __restrict__ escore, unsigned* __restrict__ emax,
                                   int C) {
  int t = blockIdx.x * blockDim.x + threadIdx.x;
  if (t >= E_TOT * HEADS) return;
  int e = t >> 2, h = t & 3;
  int s, d; edge_endpoints(eidx, e, s, d);
  int F = HEADS * C;
  const float* pl = xl + (long)s * F + h * C;
  const float* pr = xr + (long)d * F + h * C;
  const float* pa = att + h * C;
  float acc = 0.f;
  for (int c = 0; c < C; ++c) {
    float v = pl[c] + pr[c];
    v = v > 0.f ? v : NEG_SLOPE * v;
    acc += pa[c] * v;
  }
  escore[t] = acc;
  // monotonic flipped-uint so unsigned max == float max; init value 0 < flip(-inf)
  unsigned u = __float_as_uint(acc);
  u = (u & 0x80000000u) ? ~u : (u | 0x80000000u);
  atomicMax(&emax[d * HEADS + h], u);
}

// per (edge, head): w = exp(e - max); seg-sum into denom
__global__ void edge_softmax_kernel(float* __restrict__ escore, const unsigned* __restrict__ emax,
                                    const int* __restrict__ eidx, float* __restrict__ denom) {
  int t = blockIdx.x * blockDim.x + threadIdx.x;
  if (t >= E_TOT * HEADS) return;
  int e = t >> 2, h = t & 3;
  int s, d; edge_endpoints(eidx, e, s, d);
  unsigned u = emax[d * HEADS + h];
  float mx = (u & 0x80000000u) ? __uint_as_float(u ^ 0x80000000u)
                               : __uint_as_float(~u);
  float w = expf(escore[t] - mx);
  escore[t] = w;
  atomicAdd(&denom[d * HEADS + h], w);
}

// per (edge, head): out[dst] += (w/denom[dst]) * xl[src]
__global__ void edge_aggregate_kernel(const float* __restrict__ escore, const float* __restrict__ denom,
                                      const float* __restrict__ xl, const int* __restrict__ eidx,
                                      float* __restrict__ out, int C) {
  int t = blockIdx.x * blockDim.x + threadIdx.x;
  if (t >= E_TOT * HEADS) return;
  int e = t >> 2, h = t & 3;
  int s, d; edge_endpoints(eidx, e, s, d);
  float alpha = escore[t] / denom[d * HEADS + h];
  int F = HEADS * C;
  const float* pl = xl + (long)s * F + h * C;
  float* po = out + (long)d * F + h * C;
  for (int c = 0; c < C; ++c) atomicAdd(&po[c], alpha * pl[c]);
}

// ------------------------------------------------------------------
// elementwise / head-mean / graph-norm
// ------------------------------------------------------------------
__global__ void add_bias_kernel(float* __restrict__ x, const float* __restrict__ bias,
                                long n, int F) {
  long i = (long)blockIdx.x * blockDim.x + threadIdx.x;
  if (i < n) x[i] += bias[(int)(i % F)];
}

__global__ void head_mean_bias_kernel(const float* __restrict__ agg, const float* __restrict__ bias,
                                      float* __restrict__ out, int C) {
  long i = (long)blockIdx.x * blockDim.x + threadIdx.x;
  if (i >= (long)N_NODES * C) return;
  int n = (int)(i / C), c = (int)(i % C);
  float sum = 0.f;
  for (int h = 0; h < HEADS; ++h) sum += agg[(long)n * HEADS * C + h * C + c];
  out[i] = sum * (1.f / HEADS) + bias[c];
}

__global__ void count_nodes_kernel(const int* __restrict__ batch, float* __restrict__ counts) {
  int n = blockIdx.x * blockDim.x + threadIdx.x;
  if (n < N_NODES) atomicAdd(&counts[batch[n]], 1.f);
}

__global__ void gn_sum_kernel(const float* __restrict__ x, const int* __restrict__ batch,
                              float* __restrict__ sums, int F) {
  long i = (long)blockIdx.x * blockDim.x + threadIdx.x;
  if (i >= (long)N_NODES * F) return;
  int n = (int)(i / F), c = (int)(i % F);
  atomicAdd(&sums[batch[n] * F + c], x[i]);
}

__global__ void gn_var_kernel(const float* __restrict__ x, const int* __restrict__ batch,
                              const float* __restrict__ sums, const float* __restrict__ counts,
                              const float* __restrict__ mscale, float* __restrict__ varsums, int F) {
  long i = (long)blockIdx.x * blockDim.x + threadIdx.x;
  if (i >= (long)N_NODES * F) return;
  int n = (int)(i / F), c = (int)(i % F);
  int g = batch[n];
  float cnt = fmaxf(counts[g], 1.f);
  float h2 = x[i] - (sums[g * F + c] / cnt) * mscale[c];
  atomicAdd(&varsums[g * F + c], h2 * h2);
}

__global__ void gn_norm_relu_kernel(const float* __restrict__ x, const int* __restrict__ batch,
                                    const float* __restrict__ sums, const float* __restrict__ varsums,
                                    const float* __restrict__ counts, const float* __restrict__ mscale,
                                    const float* __restrict__ w, const float* __restrict__ b,
                                    float* __restrict__ y, int F) {
  long i = (long)blockIdx.x * blockDim.x + threadIdx.x;
  if (i >= (long)N_NODES * F) return;
  int n = (int)(i / F), c = (int)(i % F);
  int g = batch[n];
  float cnt = fmaxf(counts[g], 1.f);
  float h2 = x[i] - (sums[g * F + c] / cnt) * mscale[c];
  float v  = varsums[g * F + c] / cnt;
  float r  = h2 * rsqrtf(v + GN_EPS) * w[c] + b[c];
  y[i] = r > 0.f ? r : 0.f;
}

// ------------------------------------------------------------------
// pooling + final linear
// ------------------------------------------------------------------
__global__ void feat_sum_kernel(const float* __restrict__ h, const int* __restrict__ batch,
                                float* __restrict__ feat) {
  long i = (long)blockIdx.x * blockDim.x + threadIdx.x;
  if (i >= (long)N_NODES * 8) return;
  int n = (int)(i >> 3), c = (int)(i & 7);
  atomicAdd(&feat[batch[n] * 8 + c], h[i]);
}

__global__ void finalize_kernel(const float* __restrict__ feat, const float* __restrict__ counts,
                                const float* __restrict__ lin_w, const float* __restrict__ lin_b,
                                float* __restrict__ out) {
  int g = threadIdx.x;
  if (g >= N_GRAPHS) return;
  float f[8];
  float cnt = fmaxf(counts[g], 1.f);
  for (int c = 0; c < 8; ++c) f[c] = feat[g * 8 + c] / cnt;
  for (int j = 0; j < 4; ++j) {
    float l = lin_b[j];
    for (int c = 0; c < 8; ++c) l += f[c] * lin_w[c * 4 + j];
    out[g * 4 + j] = l;                       // logits [64,4] first
  }
  for (int c = 0; c < 8; ++c) out[N_GRAPHS * 4 + g * 8 + c] = f[c];  // then features [64,8]
}

// ------------------------------------------------------------------
// host-side helpers (all launches on `stream`, capture-safe)
// ------------------------------------------------------------------
static void gat_layer(const float* xin, int K, int C, int F,
                      const float* wl, const float* wr, const float* att,
                      const int* eidx,
                      float* xl, float* xr, float* agg,
                      float* escore, unsigned* emax, float* denom,
                      hipStream_t s) {
  long waves = (long)(N_NODES / 16) * (F / 16);
  unsigned gb = gdim(waves * 32, 256);
  gemm_wmma_f32<<<gb, 256, 0, s>>>(xin, wl, xl, N_NODES, F, K);
  gemm_wmma_f32<<<gb, 256, 0, s>>>(xin, wr, xr, N_NODES, F, K);
  zero_buf(emax,  (long)N_NODES * HEADS, s);
  zero_buf(denom, (long)N_NODES * HEADS, s);
  long th = (long)E_TOT * HEADS;
  edge_scores_kernel  <<<gdim(th, 256), 256, 0, s>>>(xl, xr, att, eidx, escore, emax, C);
  edge_softmax_kernel <<<gdim(th, 256), 256, 0, s>>>(escore, emax, eidx, denom);
  zero_buf(agg, (long)N_NODES * F, s);
  edge_aggregate_kernel<<<gdim(th, 256), 256, 0, s>>>(escore, denom, xl, eidx, agg, C);
}

static void graph_norm_relu(const float* xin, const int* batch,
                            const float* w, const float* b, const float* ms,
                            const float* counts, float* sums, float* varsums,
                            float* y, int F, hipStream_t s) {
  zero_buf(sums,    (long)N_GRAPHS * F, s);
  zero_buf(varsums, (long)N_GRAPHS * F, s);
  long n = (long)N_NODES * F;
  gn_sum_kernel      <<<gdim(n, 256), 256, 0, s>>>(xin, batch, sums, F);
  gn_var_kernel      <<<gdim(n, 256), 256, 0, s>>>(xin, batch, sums, counts, ms, varsums, F);
  gn_norm_relu_kernel<<<gdim(n, 256), 256, 0, s>>>(xin, batch, sums, varsums, counts, ms, w, b, y, F);
}

extern "C" void kernel_launch(void* const* d_in, const int* in_sizes, int n_in,
                              void* d_out, int out_size, void* d_ws, size_t ws_size,
                              hipStream_t stream) {
  const float* x     = (const float*)d_in[0];
  const int*   eidx  = (const int*)  d_in[1];
  const int*   batch = (const int*)  d_in[2];
  const float* w_l1 = (const float*)d_in[3],  *w_r1 = (const float*)d_in[4];
  const float* att1 = (const float*)d_in[5],  *b1   = (const float*)d_in[6];
  const float* gn1w = (const float*)d_in[7],  *gn1b = (const float*)d_in[8],  *gn1s = (const float*)d_in[9];
  const float* w_l2 = (const float*)d_in[10], *w_r2 = (const float*)d_in[11];
  const float* att2 = (const float*)d_in[12], *b2   = (const float*)d_in[13];
  const float* gn2w = (const float*)d_in[14], *gn2b = (const float*)d_in[15], *gn2s = (const float*)d_in[16];
  const float* w_l3 = (const float*)d_in[17], *w_r3 = (const float*)d_in[18];
  const float* att3 = (const float*)d_in[19], *b3   = (const float*)d_in[20];
  const float* gn3w = (const float*)d_in[21], *gn3b = (const float*)d_in[22], *gn3s = (const float*)d_in[23];
  const float* lin_w = (const float*)d_in[24], *lin_b = (const float*)d_in[25];

  float* ws = (float*)d_ws;
  size_t off = 0;
  auto carve = [&](size_t n) { float* p = ws + off; off += n; return p; };
  const size_t BIG = (size_t)N_NODES * 128;
  float* P0 = carve(BIG);   // xl
  float* P1 = carve(BIG);   // xr
  float* P2 = carve(BIG);   // agg
  float* P3 = carve(BIG);   // layer-1 output
  float* P4 = carve(BIG);   // layer-2 output
  float*    escore = carve((size_t)E_TOT * HEADS);
  unsigned* emax   = (unsigned*)carve((size_t)N_NODES * HEADS);
  float*    denom  = carve((size_t)N_NODES * HEADS);
  float* sums    = carve((size_t)N_GRAPHS * 128);
  float* varsums = carve((size_t)N_GRAPHS * 128);
  float* counts  = carve(N_GRAPHS);
  float* feat    = carve((size_t)N_GRAPHS * 8);
  float* hm      = carve((size_t)N_NODES * 8);   // layer-3 head-mean
  float* y3      = carve((size_t)N_NODES * 8);   // layer-3 output

  // per-graph node counts (recomputed every call; graph-capture safe)
  zero_buf(counts, N_GRAPHS, stream);
  count_nodes_kernel<<<gdim(N_NODES, 256), 256, 0, stream>>>(batch, counts);

  // Layer 1: 128 -> 4x32 concat (F=128)
  gat_layer(x, 128, 32, 128, w_l1, w_r1, att1, eidx, P0, P1, P2, escore, emax, denom, stream);
  add_bias_kernel<<<gdim((long)N_NODES * 128, 256), 256, 0, stream>>>(P2, b1, (long)N_NODES * 128, 128);
  graph_norm_relu(P2, batch, gn1w, gn1b, gn1s, counts, sums, varsums, P3, 128, stream);

  // Layer 2: 128 -> 4x16 concat (F=64)
  gat_layer(P3, 128, 16, 64, w_l2, w_r2, att2, eidx, P0, P1, P2, escore, emax, denom, stream);
  add_bias_kernel<<<gdim((long)N_NODES * 64, 256), 256, 0, stream>>>(P2, b2, (long)N_NODES * 64, 64);
  graph_norm_relu(P2, batch, gn2w, gn2b, gn2s, counts, sums, varsums, P4, 64, stream);

  // Layer 3: 64 -> 4x8 mean-over-heads (F=32 -> 8)
  gat_layer(P4, 64, 8, 32, w_l3, w_r3, att3, eidx, P0, P1, P2, escore, emax, denom, stream);
  head_mean_bias_kernel<<<gdim((long)N_NODES * 8, 256), 256, 0, stream>>>(P2, b3, hm, 8);
  graph_norm_relu(hm, batch, gn3w, gn3b, gn3s, counts, sums, varsums, y3, 8, stream);

  // global mean pool + linear head; write (logits, features) into d_out
  zero_buf(feat, (long)N_GRAPHS * 8, stream);
  feat_sum_kernel<<<gdim((long)N_NODES * 8, 256), 256, 0, stream>>>(y3, batch, feat);
  finalize_kernel<<<1, 64, 0, stream>>>(feat, counts, lin_w, lin_b, (float*)d_out);
}